// Deepseekv3MoE_25572235280536
// MI455X (gfx1250) — compile-verified
//
#include <hip/hip_runtime.h>

// ---------------- problem constants ----------------
#define TK   256     // tokens
#define HD   1024    // hidden
#define NE   64      // experts
#define ID   512     // routed intermediate
#define SID  2048    // shared intermediate
#define TOPK 6
#define NGRP 8
#define GSZ  8       // NE / NGRP
#define TOPG 4
#define RSCALE 2.5f

typedef __attribute__((ext_vector_type(2))) float v2f;
typedef __attribute__((ext_vector_type(4))) float v4f;
typedef __attribute__((ext_vector_type(8))) float v8f;

__device__ __forceinline__ v8f wmma4(v2f a, v2f b, v8f c) {
  // D = A(16x4,f32) * B(4x16,f32) + C(16x16,f32)
  return __builtin_amdgcn_wmma_f32_16x16x4_f32(false, a, false, b, (short)0, c,
                                               false, false);
}

// sigmoid via v_exp + v_rcp (avoids the IEEE-divide Newton chain)
__device__ __forceinline__ float sigmoid_f(float v) {
  return __builtin_amdgcn_rcpf(1.0f + __expf(-v));
}

// ---------------- kernel 0: zero expert counters ----------------
__global__ void init_counts(int* __restrict__ ecount) {
  if (threadIdx.x < NE) ecount[threadIdx.x] = 0;
}

// ---------------- kernel 1: gate logits + noaux_tc routing ----------------
// grid = T blocks, 64 threads (thread e -> expert e)
__global__ void gate_route(const float* __restrict__ x,
                           const float* __restrict__ gw,
                           const float* __restrict__ ebias,
                           int* __restrict__ ecount,
                           int* __restrict__ elist,
                           float* __restrict__ wslot) {
  const int t = blockIdx.x;
  const int e = threadIdx.x;  // 0..63
  __shared__ float sc[NE];    // sigmoid(logit)
  __shared__ float swb[NE];   // score + bias
  __shared__ float mv[NE];    // masked values for top-k

  const v4f* xr = (const v4f*)(x + (long long)t * HD);
  const v4f* gr = (const v4f*)(gw + (long long)e * HD);
  float acc = 0.0f;
  for (int h = 0; h < HD / 4; ++h) {
    v4f a = xr[h];
    v4f b = gr[h];
    acc += a[0] * b[0] + a[1] * b[1] + a[2] * b[2] + a[3] * b[3];
  }
  // exact sigmoid here (routing decisions feed normalization; keep it tight)
  float s = 1.0f / (1.0f + __expf(-acc));
  sc[e] = s;
  swb[e] = s + ebias[e];
  __syncthreads();

  if (e == 0) {
    // group score = sum of top-2 (score+bias) in each group of 8
    float gsc[NGRP];
    for (int g = 0; g < NGRP; ++g) {
      float m1 = -1e30f, m2 = -1e30f;
      for (int j = 0; j < GSZ; ++j) {
        float v = swb[g * GSZ + j];
        if (v > m1) { m2 = m1; m1 = v; }
        else if (v > m2) { m2 = v; }
      }
      gsc[g] = m1 + m2;
    }
    // top-4 groups (ties -> lowest index, matching jax top_k)
    unsigned gmask = 0u;
    for (int r = 0; r < TOPG; ++r) {
      int best = 0; float bv = -1e30f;
      for (int g = 0; g < NGRP; ++g)
        if (!((gmask >> g) & 1u) && gsc[g] > bv) { bv = gsc[g]; best = g; }
      gmask |= (1u << best);
    }
    // mask: swb * smask (masked -> 0, matching jax)
    for (int i = 0; i < NE; ++i)
      mv[i] = ((gmask >> (i / GSZ)) & 1u) ? swb[i] : 0.0f;
    // top-6 experts; combine weight = score / sum(score) * 2.5
    int   tidx[TOPK];
    float sval[TOPK];
    float ssum = 0.0f;
    for (int r = 0; r < TOPK; ++r) {
      int best = 0; float bv = -1e30f;
      for (int i = 0; i < NE; ++i)
        if (mv[i] > bv) { bv = mv[i]; best = i; }
      mv[best] = -1e30f;
      tidx[r] = best;
      sval[r] = sc[best];
      ssum += sc[best];
    }
    float inv = RSCALE / (ssum + 1e-20f);
    for (int r = 0; r < TOPK; ++r) {
      int ex = tidx[r];
      int pos = atomicAdd(&ecount[ex], 1);
      elist[ex * TK + pos] = t * TOPK + r;  // entry id encodes (token, slot)
      wslot[t * TOPK + r] = sval[r] * inv;
    }
  }
}

// ---------------- WMMA gate+up tile core ----------------
// Computes a [16 x 64] tile of silu(X*Wg^T) * (X*Wu^T), row-scaled, stored to out.
// A rows come either from a gather list (HAS_LIST) or densely.
template <bool HAS_LIST>
__device__ __forceinline__ void gateup_tile(
    const float* __restrict__ x, int lda,
    const float* __restrict__ Wg, const float* __restrict__ Wu, int ldw, int K,
    float* __restrict__ out, int ldo,
    const int* __restrict__ list, int nrows,
    const float* __restrict__ wscale, int col0, int m0) {
  const int lane = threadIdx.x & 31;
  const int half = lane >> 4;   // 0 or 1: selects K-pair
  const int l15  = lane & 15;

  // A-lane row (clamped; invalid rows masked at store)
  int arow = m0 + l15;
  if (arow > nrows - 1) arow = nrows - 1;
  int aent = HAS_LIST ? list[arow] : arow;
  long long arowg = HAS_LIST ? (long long)(aent / TOPK) : (long long)aent;
  const float* aptr = x + arowg * lda + half * 2;

  const float* bg[4];
  const float* bu[4];
#pragma unroll
  for (int c = 0; c < 4; ++c) {
    long long wr = (long long)(col0 + c * 16 + l15);
    bg[c] = Wg + wr * ldw + half * 2;
    bu[c] = Wu + wr * ldw + half * 2;
  }

  v8f accg[4] = {v8f{}, v8f{}, v8f{}, v8f{}};
  v8f accu[4] = {v8f{}, v8f{}, v8f{}, v8f{}};
  for (int k = 0; k < K; k += 4) {
    v2f a = *(const v2f*)(aptr + k);
#pragma unroll
    for (int c = 0; c < 4; ++c) {
      v2f b = *(const v2f*)(bg[c] + k);
      accg[c] = wmma4(a, b, accg[c]);
    }
#pragma unroll
    for (int c = 0; c < 4; ++c) {
      v2f b = *(const v2f*)(bu[c] + k);
      accu[c] = wmma4(a, b, accu[c]);
    }
  }

  // epilogue: act = silu(g) * u, scale row by combine weight, store
#pragma unroll
  for (int j = 0; j < 8; ++j) {
    int row = m0 + j + half * 8;
    if (row < nrows) {
      int ent = HAS_LIST ? list[row] : row;
      float w = HAS_LIST ? wscale[ent] : 1.0f;
      long long ob = (long long)ent * ldo + col0 + l15;
#pragma unroll
      for (int c = 0; c < 4; ++c) {
        float g = accg[c][j];
        float u = accu[c][j];
        out[ob + c * 16] = g * sigmoid_f(g) * u * w;
      }
    }
  }
}

// ---------------- kernel 2: routed gate+up (gathered) ----------------
// grid = (ID/64, NE), block = 32 (one wave)
__global__ void routed_gateup(const float* __restrict__ x,
                              const float* __restrict__ w1,
                              const float* __restrict__ w3,
                              float* __restrict__ act,
                              const int* __restrict__ ecount,
                              const int* __restrict__ elist,
                              const float* __restrict__ wslot) {
  const int e = blockIdx.y;
  const int col0 = blockIdx.x * 64;
  const int n = ecount[e];
  if (n <= 0) return;
  const int* list = elist + e * TK;
  const float* Wg = w1 + (long long)e * ID * HD;
  const float* Wu = w3 + (long long)e * ID * HD;
  for (int m0 = 0; m0 < n; m0 += 16)
    gateup_tile<true>(x, HD, Wg, Wu, HD, HD, act, ID, list, n, wslot, col0, m0);
}

// ---------------- kernel 3: routed down-proj (gathered GEMM) ----------------
// grid = (HD/64, NE), block = 32
__global__ void routed_down(const float* __restrict__ act,
                            const float* __restrict__ w2,
                            float* __restrict__ partial,
                            const int* __restrict__ ecount,
                            const int* __restrict__ elist) {
  const int e = blockIdx.y;
  const int col0 = blockIdx.x * 64;
  const int n = ecount[e];
  if (n <= 0) return;
  const int* list = elist + e * TK;
  const float* W = w2 + (long long)e * HD * ID;
  const int lane = threadIdx.x & 31;
  const int half = lane >> 4;
  const int l15  = lane & 15;

  for (int m0 = 0; m0 < n; m0 += 16) {
    int arow = m0 + l15;
    if (arow > n - 1) arow = n - 1;
    int aent = list[arow];
    const float* aptr = act + (long long)aent * ID + half * 2;
    const float* bp[4];
#pragma unroll
    for (int c = 0; c < 4; ++c)
      bp[c] = W + (long long)(col0 + c * 16 + l15) * ID + half * 2;

    v8f acc[4] = {v8f{}, v8f{}, v8f{}, v8f{}};
    for (int k = 0; k < ID; k += 4) {
      v2f a = *(const v2f*)(aptr + k);
#pragma unroll
      for (int c = 0; c < 4; ++c) {
        v2f b = *(const v2f*)(bp[c] + k);
        acc[c] = wmma4(a, b, acc[c]);
      }
    }
#pragma unroll
    for (int j = 0; j < 8; ++j) {
      int row = m0 + j + half * 8;
      if (row < n) {
        int ent = list[row];
        long long ob = (long long)ent * HD + col0 + l15;
#pragma unroll
        for (int c = 0; c < 4; ++c) partial[ob + c * 16] = acc[c][j];
      }
    }
  }
}

// ---------------- kernel 4: shared expert gate+up (dense) ----------------
// grid = (SID/64, TK/16), block = 32
__global__ void shared_gateup(const float* __restrict__ x,
                              const float* __restrict__ wsg,
                              const float* __restrict__ wsu,
                              float* __restrict__ sact) {
  const int col0 = blockIdx.x * 64;
  const int m0 = blockIdx.y * 16;
  gateup_tile<false>(x, HD, wsg, wsu, HD, HD, sact, SID, nullptr, TK, nullptr,
                     col0, m0);
}

// ---------------- kernel 5: shared down-proj + combine routed partials ----
// grid = (HD/64, TK/16), block = 32
__global__ void shared_down_combine(const float* __restrict__ sact,
                                    const float* __restrict__ wsd,
                                    const float* __restrict__ partial,
                                    float* __restrict__ out) {
  const int col0 = blockIdx.x * 64;
  const int m0 = blockIdx.y * 16;
  const int lane = threadIdx.x & 31;
  const int half = lane >> 4;
  const int l15  = lane & 15;

  const float* aptr = sact + (long long)(m0 + l15) * SID + half * 2;
  const float* bp[4];
#pragma unroll
  for (int c = 0; c < 4; ++c)
    bp[c] = wsd + (long long)(col0 + c * 16 + l15) * SID + half * 2;

  v8f acc[4] = {v8f{}, v8f{}, v8f{}, v8f{}};
  for (int k = 0; k < SID; k += 4) {
    v2f a = *(const v2f*)(aptr + k);
#pragma unroll
    for (int c = 0; c < 4; ++c) {
      v2f b = *(const v2f*)(bp[c] + k);
      acc[c] = wmma4(a, b, acc[c]);
    }
  }
#pragma unroll
  for (int j = 0; j < 8; ++j) {
    int row = m0 + j + half * 8;
#pragma unroll
    for (int c = 0; c < 4; ++c) {
      int col = col0 + c * 16 + l15;
      const float* pp = partial + (long long)row * TOPK * HD + col;
      float r = pp[0] + pp[HD] + pp[2 * HD] + pp[3 * HD] + pp[4 * HD] + pp[5 * HD];
      out[(long long)row * HD + col] = acc[c][j] + r;
    }
  }
}

// ---------------- launch ----------------
extern "C" void kernel_launch(void* const* d_in, const int* in_sizes, int n_in,
                              void* d_out, int out_size, void* d_ws,
                              size_t ws_size, hipStream_t stream) {
  const float* x       = (const float*)d_in[0];  // [T,H]
  const float* gate_w  = (const float*)d_in[1];  // [E,H]
  const float* e_bias  = (const float*)d_in[2];  // [E]
  const float* w1      = (const float*)d_in[3];  // [E,I,H]
  const float* w3      = (const float*)d_in[4];  // [E,I,H]
  const float* w2      = (const float*)d_in[5];  // [E,H,I]
  const float* ws_gate = (const float*)d_in[6];  // [SI,H]
  const float* ws_up   = (const float*)d_in[7];  // [SI,H]
  const float* ws_down = (const float*)d_in[8];  // [H,SI]
  float* out = (float*)d_out;                    // [T,H]

  // workspace carve-up
  char* p = (char*)d_ws;
  int*   ecount  = (int*)p;    p += 256;                                    // [E]
  int*   elist   = (int*)p;    p += (size_t)NE * TK * sizeof(int);          // [E,T]
  float* wslot   = (float*)p;  p += (size_t)TK * TOPK * sizeof(float);      // [T,6]
  float* act     = (float*)p;  p += (size_t)TK * TOPK * ID * sizeof(float); // [T*6,I]
  float* partial = (float*)p;  p += (size_t)TK * TOPK * HD * sizeof(float); // [T*6,H]
  float* sact    = (float*)p;  /* [T,SI] */                                 // 2MB

  init_counts<<<1, 64, 0, stream>>>(ecount);
  gate_route<<<TK, 64, 0, stream>>>(x, gate_w, e_bias, ecount, elist, wslot);

  dim3 g2(ID / 64, NE);
  routed_gateup<<<g2, 32, 0, stream>>>(x, w1, w3, act, ecount, elist, wslot);

  dim3 g3(HD / 64, NE);
  routed_down<<<g3, 32, 0, stream>>>(act, w2, partial, ecount, elist);

  dim3 g4(SID / 64, TK / 16);
  shared_gateup<<<g4, 32, 0, stream>>>(x, ws_gate, ws_up, sact);

  dim3 g5(HD / 64, TK / 16);
  shared_down_combine<<<g5, 32, 0, stream>>>(sact, ws_down, partial, out);
}